// GroupTopk_65154653880340
// MI455X (gfx1250) — compile-verified
//
#include <hip/hip_runtime.h>

// GroupTopk: x[16,512,64,64] f32, groups of 32 channels, top-4 per pixel-group,
// out = x + w[g] (32x4) @ topk (4).  Memory-bound: 256 MiB traffic ~ 11 us @ 23.3 TB/s.
// WMMA f32 16x16x4 computes D = W_tile(16x4) x TopK(4x16 pixels) + X_tile(16x16),
// i.e. the grouped 1x1 conv AND the residual add in one matrix op.

typedef __attribute__((ext_vector_type(2))) float v2f;
typedef __attribute__((ext_vector_type(8))) float v8f;

#define HW      4096          // 64*64
#define CSTRIDE 4096          // per-channel pixel stride (floats)

__device__ __forceinline__ void topk_insert(float v, float& t0, float& t1, float& t2, float& t3) {
    // branchless sorted-descending top-4 insertion
    float m;
    m = fminf(v, t0); t0 = fmaxf(v, t0); v = m;
    m = fminf(v, t1); t1 = fmaxf(v, t1); v = m;
    m = fminf(v, t2); t2 = fmaxf(v, t2); v = m;
    t3 = fmaxf(v, t3);
}

__device__ __forceinline__ float swap16(float x) {
    // ds_swizzle SWAPX16: and=0x1f, or=0, xor=0x10  -> offset 0x401f
    return __int_as_float(__builtin_amdgcn_ds_swizzle(__float_as_int(x), 0x401f));
}

__global__ __launch_bounds__(256) void grouptopk_wmma(
    const float* __restrict__ x, const float* __restrict__ w, float* __restrict__ out)
{
    const int lane = threadIdx.x & 31;
    const int wave = threadIdx.x >> 5;
    const int half = lane >> 4;     // 0: lanes 0-15, 1: lanes 16-31
    const int col  = lane & 15;     // pixel within tile / M row index

    // tile id: 65536 waves total; 256 pixel-tiles per (n,g)
    const int tile = blockIdx.x * 8 + wave;
    const int hw16 = tile & 255;          // which 16-pixel chunk of the 4096 pixels
    const int ng   = tile >> 8;
    const int g    = ng & 15;
    const int n    = ng >> 4;

    const int base = n * (512 * HW) + g * (32 * HW) + hw16 * 16 + col;
    const float* __restrict__ xb = x + base;
    float* __restrict__ ob = out + base;

    // C/D tile layout: VGPR v holds channel (v + 8*half); tile1 adds +16 channels.
    const int choff = half * 8 * CSTRIDE;

    // ---- load x tiles (each b32 load = two full 64B segments) ----
    v8f c0, c1;
    #pragma unroll
    for (int v = 0; v < 8; ++v) {
        c0[v] = xb[choff + v * CSTRIDE];            // channels 0..15  (as ch = v + 8*half)
        c1[v] = xb[choff + (v + 16) * CSTRIDE];     // channels 16..31
    }

    // ---- per-lane top-4 over this lane's 16 channel values ----
    float t0 = -INFINITY, t1 = -INFINITY, t2 = -INFINITY, t3 = -INFINITY;
    #pragma unroll
    for (int v = 0; v < 8; ++v) topk_insert(c0[v], t0, t1, t2, t3);
    #pragma unroll
    for (int v = 0; v < 8; ++v) topk_insert(c1[v], t0, t1, t2, t3);

    // ---- merge with partner half-lane (same pixel, other 16 channels) ----
    float s0 = swap16(t0), s1 = swap16(t1), s2 = swap16(t2), s3 = swap16(t3);
    topk_insert(s0, t0, t1, t2, t3);
    topk_insert(s1, t0, t1, t2, t3);
    topk_insert(s2, t0, t1, t2, t3);
    topk_insert(s3, t0, t1, t2, t3);
    // Both half-lanes now hold pixel 'col' top-4 (descending t0..t3).

    // ---- B matrix (4x16): VGPR0 = K0 (lanes 0-15) / K2 (lanes 16-31); VGPR1 = K1/K3 ----
    v2f bm;
    bm.x = half ? t2 : t0;
    bm.y = half ? t3 : t1;

    // ---- A matrices (16x4) from w[g][o][k]: lane M = out channel, half selects K pair ----
    const float* __restrict__ wg = w + g * (32 * 4);
    const int k0 = half ? 2 : 0;
    v2f a0, a1;
    a0.x = wg[col * 4 + k0];
    a0.y = wg[col * 4 + k0 + 1];
    a1.x = wg[(col + 16) * 4 + k0];
    a1.y = wg[(col + 16) * 4 + k0 + 1];

    // ---- D = A x B + C : grouped 1x1 conv + residual in one WMMA each ----
    v8f d0 = __builtin_amdgcn_wmma_f32_16x16x4_f32(false, a0, false, bm, (short)0, c0, false, false);
    v8f d1 = __builtin_amdgcn_wmma_f32_16x16x4_f32(false, a1, false, bm, (short)0, c1, false, false);

    // ---- store (mirrors the load pattern: two 64B segments per store) ----
    #pragma unroll
    for (int v = 0; v < 8; ++v) {
        ob[choff + v * CSTRIDE]        = d0[v];
        ob[choff + (v + 16) * CSTRIDE] = d1[v];
    }
}

extern "C" void kernel_launch(void* const* d_in, const int* in_sizes, int n_in,
                              void* d_out, int out_size, void* d_ws, size_t ws_size,
                              hipStream_t stream) {
    const float* x = (const float*)d_in[0];   // [16,512,64,64]
    const float* w = (const float*)d_in[1];   // [16,32,4]
    float* out = (float*)d_out;               // [16,512,64,64]

    // 16 n * 16 g * 256 pixel-tiles = 65536 waves = 8192 blocks of 8 waves
    dim3 grid(8192), block(256);
    hipLaunchKernelGGL(grouptopk_wmma, grid, block, 0, stream, x, w, out);
}